// MolAttention_22222160790046
// MI455X (gfx1250) — compile-verified
//
#include <hip/hip_runtime.h>

#define B_ 64
#define T_ 8192
#define D_ 512
#define Q_ 128
#define U_ 8
#define K_ 5
#define EPS_ 1e-8f
#define CHUNK_ 128

typedef float v2f __attribute__((ext_vector_type(2)));
typedef float v8f __attribute__((ext_vector_type(8)));

// ---------------- zero output (we accumulate via atomics) ----------------
__global__ void mol_zero(float* __restrict__ out, int n) {
    int i = blockIdx.x * blockDim.x + threadIdx.x;
    if (i < n) out[i] = 0.0f;
}

// ---------------- tiny MLP via fp32 WMMA: query[64,128] @ W1[128,8] ------
// 4 waves, each owns a 16-batch M-tile; K=128 swept in 32 steps of 4.
// A 16x4 layout: lanes 0-15 -> M, K split lo/hi across lane halves, 2 VGPRs.
// B columns >= U_ are zero-padded branchlessly (clamped address * lane mask)
// so the inner loop has no exec-mask manipulation around its loads.
__global__ __launch_bounds__(128) void mol_params(
    const float* __restrict__ query, const float* __restrict__ attst,
    const float* __restrict__ W1, const float* __restrict__ b1,
    const float* __restrict__ W2, float* __restrict__ params) {
    __shared__ float hsh[B_][U_ + 1];   // +1 pad to dodge bank conflicts
    const int tid  = threadIdx.x;
    const int lane = tid & 31;
    const int wave = tid >> 5;
    const int b0   = wave * 16;             // batch tile base
    const int half = lane >> 4;             // 0: K={0,1}, 1: K={2,3}
    const int mn   = lane & 15;             // M index for A, N index for B
    const int ncl  = mn & (U_ - 1);         // clamped (always valid) B column
    const float bmask = (mn < U_) ? 1.0f : 0.0f;  // zero-pad N >= 8

    v8f c = {};
    #pragma unroll 4
    for (int k0 = 0; k0 < Q_; k0 += 4) {
        const int kb = k0 + half * 2;
        v2f a, bm;
        a.x  = query[(b0 + mn) * Q_ + kb];
        a.y  = query[(b0 + mn) * Q_ + kb + 1];
        bm.x = W1[kb * U_ + ncl]       * bmask;   // branchless zero-pad
        bm.y = W1[(kb + 1) * U_ + ncl] * bmask;
        c = __builtin_amdgcn_wmma_f32_16x16x4_f32(
                /*neg_a=*/false, a, /*neg_b=*/false, bm,
                /*c_mod=*/(short)0, c, /*reuse_a=*/false, /*reuse_b=*/false);
    }
    // C/D layout: VGPR r -> M=r (lanes 0-15) / M=r+8 (lanes 16-31), N=lane&15
    #pragma unroll
    for (int r = 0; r < 8; ++r) {
        int m = r + half * 8;
        if (mn < U_) hsh[b0 + m][mn] = c[r];
    }
    __syncthreads();

    if (tid < B_) {
        const int b = tid;
        float h[U_];
        #pragma unroll
        for (int u = 0; u < U_; ++u) h[u] = tanhf(hsh[b][u] + b1[u]);
        float proj[3 * K_];
        #pragma unroll
        for (int j = 0; j < 3 * K_; ++j) {
            float s = 0.0f;
            #pragma unroll
            for (int u = 0; u < U_; ++u) s = fmaf(h[u], W2[u * (3 * K_) + j], s);
            proj[j] = s;
        }
        // alpha = softmax(proj[10..14])
        float amax = proj[2 * K_];
        #pragma unroll
        for (int k = 1; k < K_; ++k) amax = fmaxf(amax, proj[2 * K_ + k]);
        float ae[K_], asum = 0.0f;
        #pragma unroll
        for (int k = 0; k < K_; ++k) { ae[k] = __expf(proj[2 * K_ + k] - amax); asum += ae[k]; }
        float inv_asum = 1.0f / asum;
        float* p = params + b * 16;
        #pragma unroll
        for (int k = 0; k < K_; ++k) {
            p[k]          = ae[k] * inv_asum;                      // alpha
            p[K_ + k]     = __expf(proj[k]) + attst[b * K_ + k];   // kappa
            p[2 * K_ + k] = 1.0f / (__expf(proj[K_ + k]) + EPS_);  // 1/(beta+eps)
        }
        p[15] = 0.0f;
    }
}

__device__ __forceinline__ float sigmoidf_(float x) {
    return 1.0f / (1.0f + __expf(-x));
}

// ---------------- bandwidth-bound stream: scores (.) values -------------
// grid(64 batches, 64 chunks of 128 t's), 128 threads; each thread owns a
// float4 column slice (4*128 = 512 = D). Coalesced 2KB rows, atomics to out.
__global__ __launch_bounds__(CHUNK_) void mol_main(
    const float* __restrict__ values, const float* __restrict__ params,
    float* __restrict__ out) {
    const int b   = blockIdx.x;
    const int t0  = blockIdx.y * CHUNK_;
    const int tid = threadIdx.x;
    __shared__ float s_sc[CHUNK_];
    __shared__ float s_red[CHUNK_];

    const float* p = params + b * 16;
    {
        float j  = (float)(t0 + tid + 1);
        float sc = 0.0f;
        #pragma unroll
        for (int k = 0; k < K_; ++k) {
            float al = p[k], kap = p[K_ + k], ib = p[2 * K_ + k];
            sc += al * (sigmoidf_((j + 0.5f - kap) * ib) -
                        sigmoidf_((j - 0.5f - kap) * ib));
        }
        s_sc[tid] = sc;
    }
    __syncthreads();

    const float4* v = (const float4*)(values + ((size_t)b * T_ + t0) * D_) + tid;
    float4 acc = make_float4(0.f, 0.f, 0.f, 0.f);
    #pragma unroll 4
    for (int tt = 0; tt < CHUNK_; ++tt) {
        float4 x = v[(size_t)tt * (D_ / 4)];
        float  s = s_sc[tt];
        acc.x = fmaf(s, x.x, acc.x);
        acc.y = fmaf(s, x.y, acc.y);
        acc.z = fmaf(s, x.z, acc.z);
        acc.w = fmaf(s, x.w, acc.w);
    }
    float* o = out + b * D_ + tid * 4;
    atomicAdd(o + 0, acc.x);
    atomicAdd(o + 1, acc.y);
    atomicAdd(o + 2, acc.z);
    atomicAdd(o + 3, acc.w);

    // loss = sum(output): block-reduce partial, one atomic per block
    s_red[tid] = acc.x + acc.y + acc.z + acc.w;
    __syncthreads();
    for (int s = CHUNK_ / 2; s > 0; s >>= 1) {
        if (tid < s) s_red[tid] += s_red[tid + s];
        __syncthreads();
    }
    if (tid == 0) atomicAdd(out + B_ * D_, s_red[0]);
}

extern "C" void kernel_launch(void* const* d_in, const int* in_sizes, int n_in,
                              void* d_out, int out_size, void* d_ws, size_t ws_size,
                              hipStream_t stream) {
    const float* query  = (const float*)d_in[0];   // [B,Q]
    const float* values = (const float*)d_in[1];   // [B,T,D]
    const float* attst  = (const float*)d_in[2];   // [B,1,K]
    const float* W1     = (const float*)d_in[3];   // [Q,U]
    const float* b1     = (const float*)d_in[4];   // [U]
    const float* W2     = (const float*)d_in[5];   // [U,3K]
    float* out    = (float*)d_out;                 // [B*D] ++ [loss]
    float* params = (float*)d_ws;                  // [B,16]: alpha|kappa|inv_beta

    mol_zero<<<(out_size + 255) / 256, 256, 0, stream>>>(out, out_size);
    mol_params<<<1, 128, 0, stream>>>(query, attst, W1, b1, W2, params);
    dim3 grid(B_, T_ / CHUNK_);
    mol_main<<<grid, CHUNK_, 0, stream>>>(values, params, out);
}